// MonotonicAttention_8375186227936
// MI455X (gfx1250) — compile-verified
//
#include <hip/hip_runtime.h>
#include <hip/hip_bf16.h>

typedef __attribute__((ext_vector_type(2))) float v2f;
typedef __attribute__((ext_vector_type(8))) float v8f;

#define B_   4
#define TGT_ 256
#define SRC_ 256
#define SS_  512
#define TS_  256

#define ESTRIDE 260   // LDS row stride: 16B-aligned rows -> ds_load_b128 / async b128

__device__ __forceinline__ float fexp2(float x) { return __builtin_amdgcn_exp2f(x); }
__device__ __forceinline__ float flog2(float x) { return __builtin_amdgcn_logf(x); }
__device__ __forceinline__ float frcp (float x) { return __builtin_amdgcn_rcpf(x); }

// tanh(x) = 1 - 2/(exp(2x)+1) ; exp(2x) = exp2(2*log2(e)*x)
__device__ __forceinline__ float ftanh(float x) {
    return 1.0f - 2.0f * frcp(1.0f + fexp2(2.8853900817779268f * x));
}
// sigmoid(x) = 1/(1+exp(-x))
__device__ __forceinline__ float fsigmoid(float x) {
    return frcp(1.0f + fexp2(-1.4426950408889634f * x));
}

// 32-bit LDS byte offset of a __shared__ object (generic -> addrspace(3) cast)
__device__ __forceinline__ unsigned lds_off(const void* p) {
    return (unsigned)(unsigned long long)
           (__attribute__((address_space(3))) const char*)p;
}

// ---------------------------------------------------------------------------
// Batched f32 WMMA GEMM:  C[b,m,n] = sum_k A[b,m,k]*B(k,n) + bias[n] + bias2[n]
//   B_N_MAJOR==1 : B stored as weight[N, K]   (B[k,n] = Bm[n*ldb + k])
//   B_N_MAJOR==0 : B stored as matrix[K, N]   (B[k,n] = Bm[k*ldb + n])
// One wave per 16x16 tile; K stepped by 4 via V_WMMA_F32_16X16X4_F32,
// unrolled x4 so four WMMAs per trip are in flight behind clause-grouped loads.
// ---------------------------------------------------------------------------
template <int B_N_MAJOR>
__global__ void wmma_gemm_f32(const float* __restrict__ A, const float* __restrict__ Bm,
                              const float* __restrict__ bias, const float* __restrict__ bias2,
                              float* __restrict__ C,
                              int M, int N, int K, int lda, int ldb, int ldc,
                              long sA, long sB, long sC, int total_tiles)
{
    int wid = blockIdx.x * (blockDim.x >> 5) + (threadIdx.x >> 5);
    if (wid >= total_tiles) return;
    int mt = M >> 4, nt = N >> 4;
    int per_b = mt * nt;
    int b   = wid / per_b;
    int rem = wid - b * per_b;
    int m0 = (rem / nt) << 4;
    int n0 = (rem % nt) << 4;

    const float* Ab = A + (long)b * sA;
    const float* Bb = Bm + (long)b * sB;

    int lane = threadIdx.x & 31;
    int r16  = lane & 15;               // A: row M = lane%16 ; B/C/D: col N = lane%16
    int koff = (lane >> 4) << 1;        // lanes 0-15 -> K{0,1}, lanes 16-31 -> K{2,3}
    const float* arow = Ab + (long)(m0 + r16) * lda;
    int n = n0 + r16;
    const float* brow = Bb + (long)n * ldb;   // used when B_N_MAJOR

    v8f acc = {0.f, 0.f, 0.f, 0.f, 0.f, 0.f, 0.f, 0.f};
    for (int kk = 0; kk < K; kk += 16) {
        __builtin_prefetch(arow + kk + 16, 0, 3);   // WGP-scope prefetch (all levels)
#pragma unroll
        for (int j = 0; j < 16; j += 4) {
            int k = kk + j + koff;
            v2f a = *(const v2f*)(arow + k);        // aligned 8B: global_load_b64
            v2f bv;
            if (B_N_MAJOR) {
                bv = *(const v2f*)(brow + k);       // weight row, 8B aligned
            } else {
                bv.x = Bb[(long)(k)     * ldb + n];
                bv.y = Bb[(long)(k + 1) * ldb + n];
            }
            acc = __builtin_amdgcn_wmma_f32_16x16x4_f32(
                      false, a, false, bv, (short)0, acc, false, false);
        }
    }

    float bb = (bias ? bias[n] : 0.0f) + (bias2 ? bias2[n] : 0.0f);
    int rbase = m0 + ((lane >> 4) << 3);    // D: VGPR r -> row r (lanes<16) / r+8
    float* Cb = C + (long)b * sC;
#pragma unroll
    for (int rr = 0; rr < 8; ++rr)
        Cb[(long)(rbase + rr) * ldc + n] = acc[rr] + bb;
}

// ---------------------------------------------------------------------------
// Fused energy kernel: per (b, 32x32 [t,s] tile)
//   energy[t,s] = sum_u tanh(encb[s,u] + dec[t,u]) * vw[u] + v_b + r
//   (bias[u] was pre-folded into enc by the GEMM epilogue)
//   p = sigmoid(energy + noise);  also emits log2(clip(1-p, 1e-10, 1))
// Tiles DMA'd into LDS with GLOBAL_LOAD_ASYNC_TO_LDS_B128 (ASYNCcnt), the
// v_dir-norm reduction overlaps the copies, fence = s_wait_asynccnt + barrier.
// Hot loop runs on ds_load_b128; dec/vw reads are wave-uniform broadcasts.
// ---------------------------------------------------------------------------
__global__ void energy_kernel(const float* __restrict__ enc, const float* __restrict__ dec,
                              const float* __restrict__ v_dir,
                              const float* __restrict__ v_g, const float* __restrict__ v_b,
                              const float* __restrict__ r_s, const float* __restrict__ noise,
                              float* __restrict__ p_sel, float* __restrict__ log1m_l2)
{
    __shared__ float enc_s[32 * ESTRIDE];
    __shared__ float dec_s[32 * ESTRIDE];
    __shared__ float vw_s[256];
    __shared__ float red_s[256];

    int tid = threadIdx.x;
    int blk = blockIdx.x;               // B * 8 * 8 blocks
    int b  = blk >> 6;
    int t0 = ((blk >> 3) & 7) << 5;
    int s0 = (blk & 7) << 5;

    // ---- issue async global->LDS tile copies (16B per lane per pass) ----
    {
        const float* encb = enc + (long)b * SRC_ * TS_;
        const float* decb = dec + (long)b * TGT_ * TS_;
        unsigned enc_base = lds_off(enc_s);
        unsigned dec_base = lds_off(dec_s);
        int row_in = tid >> 6;          // 4 rows per pass (256 threads * 4 floats)
        int u      = (tid & 63) << 2;
#pragma unroll
        for (int p = 0; p < 8; ++p) {
            int row = (p << 2) + row_in;
            const float* ge = encb + (long)(s0 + row) * TS_ + u;
            const float* gd = decb + (long)(t0 + row) * TS_ + u;
            unsigned lofs = (unsigned)((row * ESTRIDE + u) << 2);
            unsigned le = enc_base + lofs;
            unsigned ld = dec_base + lofs;
            asm volatile("global_load_async_to_lds_b128 %0, %1, off"
                         :: "v"(le), "v"(ge) : "memory");
            asm volatile("global_load_async_to_lds_b128 %0, %1, off"
                         :: "v"(ld), "v"(gd) : "memory");
        }
    }

    // ---- vw = g * v_dir / ||v_dir||  (overlaps the async DMA) ----
    float vd = v_dir[tid];
    red_s[tid] = vd * vd;
    __syncthreads();
    for (int off = 128; off > 0; off >>= 1) {
        if (tid < off) red_s[tid] += red_s[tid + off];
        __syncthreads();
    }
    vw_s[tid] = v_g[0] * vd * rsqrtf(red_s[0]);

    // ---- fence: our copies done, then all waves' copies visible ----
    asm volatile("s_wait_asynccnt 0x0" ::: "memory");
    __syncthreads();

    int tx = tid & 31;                  // s lane (varies within wave)
    int tg = tid >> 5;                  // t group (uniform per wave -> LDS broadcast)
    const float* es = &enc_s[tx * ESTRIDE];
    const float* d0 = &dec_s[(tg * 4 + 0) * ESTRIDE];
    const float* d1 = &dec_s[(tg * 4 + 1) * ESTRIDE];
    const float* d2 = &dec_s[(tg * 4 + 2) * ESTRIDE];
    const float* d3 = &dec_s[(tg * 4 + 3) * ESTRIDE];

    float a0 = 0.f, a1 = 0.f, a2 = 0.f, a3 = 0.f;
    for (int u = 0; u < 256; u += 4) {
        float4 e  = *(const float4*)(es + u);       // ds_load_b128
        float4 w  = *(const float4*)(vw_s + u);
        float4 p0 = *(const float4*)(d0 + u);
        float4 p1 = *(const float4*)(d1 + u);
        float4 p2 = *(const float4*)(d2 + u);
        float4 p3 = *(const float4*)(d3 + u);
        a0 += ftanh(e.x + p0.x) * w.x + ftanh(e.y + p0.y) * w.y
            + ftanh(e.z + p0.z) * w.z + ftanh(e.w + p0.w) * w.w;
        a1 += ftanh(e.x + p1.x) * w.x + ftanh(e.y + p1.y) * w.y
            + ftanh(e.z + p1.z) * w.z + ftanh(e.w + p1.w) * w.w;
        a2 += ftanh(e.x + p2.x) * w.x + ftanh(e.y + p2.y) * w.y
            + ftanh(e.z + p2.z) * w.z + ftanh(e.w + p2.w) * w.w;
        a3 += ftanh(e.x + p3.x) * w.x + ftanh(e.y + p3.y) * w.y
            + ftanh(e.z + p3.z) * w.z + ftanh(e.w + p3.w) * w.w;
    }

    float base = v_b[0] + r_s[0];
    int s = s0 + tx;
    float accs[4] = {a0, a1, a2, a3};
#pragma unroll
    for (int i = 0; i < 4; ++i) {
        int t = t0 + tg * 4 + i;
        long idx = ((long)b * TGT_ + t) * SRC_ + s;
        float p = fsigmoid(accs[i] + base + noise[idx]);
        p_sel[idx] = p;
        float om = 1.0f - p;
        om = fminf(fmaxf(om, 1e-10f), 1.0f);
        log1m_l2[idx] = flog2(om);      // log2 units; cumprod via exp2(cumsum)
    }
}

// cumprod over the tgt axis:  cp[b,t,s] = exp2( sum_{t'<=t} log2(1-p[b,t',s]) )
__global__ void cumprod_t_kernel(const float* __restrict__ log1m_l2, float* __restrict__ cp)
{
    int gid = blockIdx.x * blockDim.x + threadIdx.x;   // B*SRC threads
    int b = gid >> 8, s = gid & 255;
    float run = 0.f;
    for (int t = 0; t < TGT_; ++t) {
        long idx = ((long)b * TGT_ + t) * SRC_ + s;
        run += log1m_l2[idx];
        cp[idx] = fexp2(run);
    }
}

// Per (b,t) row: q = prev/cp ; inclusive scan over s ; probs = p*cp*scan*mask ;
// normalize by row-sum. One 256-thread block per row, scan/reduce in LDS.
__global__ void probs_kernel(const float* __restrict__ p_sel, const float* __restrict__ cp,
                             const float* __restrict__ prev, const float* __restrict__ mask,
                             float* __restrict__ probs)
{
    __shared__ float sbuf[256];
    int s  = threadIdx.x;
    int bt = blockIdx.x;                 // b*TGT + t
    int b  = bt >> 8;
    long idx = (long)bt * SRC_ + s;

    float cpv = cp[idx];
    float q = prev[idx] * frcp(cpv);
    sbuf[s] = q;
    __syncthreads();
    for (int off = 1; off < 256; off <<= 1) {         // Hillis-Steele inclusive scan
        float v = (s >= off) ? sbuf[s - off] : 0.0f;
        __syncthreads();
        sbuf[s] += v;
        __syncthreads();
    }
    float cq = sbuf[s];
    __syncthreads();

    float pr = p_sel[idx] * cpv * cq * mask[b * SRC_ + s];
    sbuf[s] = pr;
    __syncthreads();
    for (int off = 128; off > 0; off >>= 1) {
        if (s < off) sbuf[s] += sbuf[s + off];
        __syncthreads();
    }
    probs[idx] = pr * frcp(sbuf[0] + 1e-12f);
}

// ---------------------------------------------------------------------------
extern "C" void kernel_launch(void* const* d_in, const int* in_sizes, int n_in,
                              void* d_out, int out_size, void* d_ws, size_t ws_size,
                              hipStream_t stream)
{
    const float* hidden   = (const float*)d_in[0];
    const float* enc_out  = (const float*)d_in[1];
    const float* enc_mask = (const float*)d_in[2];
    const float* prev     = (const float*)d_in[3];
    const float* noise    = (const float*)d_in[4];
    const float* w_w      = (const float*)d_in[5];
    const float* w_b      = (const float*)d_in[6];
    const float* vl_w     = (const float*)d_in[7];
    const float* vl_b     = (const float*)d_in[8];
    const float* bias     = (const float*)d_in[9];
    const float* v_dir    = (const float*)d_in[10];
    const float* v_g      = (const float*)d_in[11];
    const float* v_b      = (const float*)d_in[12];
    const float* r_s      = (const float*)d_in[13];

    const long NE = (long)B_ * TGT_ * SRC_;            // 262144
    float* ws      = (float*)d_ws;
    float* enc_ws  = ws;                               // [B,S,T]  (+w_bias +bias folded)
    float* dec_ws  = ws + NE;                          // [B,T,T]
    float* psel_ws = ws + 2 * NE;                      // [B,T,S]
    float* log_ws  = ws + 3 * NE;                      // [B,T,S]
    float* cp_ws   = ws + 4 * NE;                      // [B,T,S]

    float* ctx_out   = (float*)d_out;                  // [B,T,512]
    float* probs_out = ctx_out + (long)B_ * TGT_ * SS_;// [B,T,S]

    // enc[b,s,t] = EO[b,s,:512] . W[t,:512] + w_bias[t] + bias[t]
    {
        int tiles = B_ * (SRC_ / 16) * (TS_ / 16);     // 1024
        wmma_gemm_f32<1><<<(tiles + 7) / 8, 256, 0, stream>>>(
            enc_out, w_w, w_b, bias, enc_ws,
            SRC_, TS_, SS_, SS_, SS_, TS_,
            (long)SRC_ * SS_, 0L, (long)SRC_ * TS_, tiles);
    }
    // dec[b,t,u] = H[b,t,:256] . V[u,:256] + v_lin_bias[u]
    {
        int tiles = B_ * (TGT_ / 16) * (TS_ / 16);     // 1024
        wmma_gemm_f32<1><<<(tiles + 7) / 8, 256, 0, stream>>>(
            hidden, vl_w, vl_b, nullptr, dec_ws,
            TGT_, TS_, TS_, TS_, TS_, TS_,
            (long)TGT_ * TS_, 0L, (long)TGT_ * TS_, tiles);
    }

    energy_kernel<<<B_ * 8 * 8, 256, 0, stream>>>(
        enc_ws, dec_ws, v_dir, v_g, v_b, r_s, noise, psel_ws, log_ws);

    cumprod_t_kernel<<<B_, 256, 0, stream>>>(log_ws, cp_ws);

    probs_kernel<<<B_ * TGT_, 256, 0, stream>>>(psel_ws, cp_ws, prev, enc_mask, probs_out);

    // context[b,t,e] = probs[b,t,:256] . EO[b,:256,e]
    {
        int tiles = B_ * (TGT_ / 16) * (SS_ / 16);     // 2048
        wmma_gemm_f32<0><<<(tiles + 7) / 8, 256, 0, stream>>>(
            probs_out, enc_out, nullptr, nullptr, ctx_out,
            TGT_, SS_, SRC_, SRC_, SS_, SS_,
            (long)TGT_ * SRC_, (long)SRC_ * SS_, (long)TGT_ * SS_, tiles);
    }
}